// RiemannSpectralController_2980707303756
// MI455X (gfx1250) — compile-verified
//
#include <hip/hip_runtime.h>
#include <math.h>

// MI455X / gfx1250, wave32. Compile: hipcc --offload-arch=gfx1250 -O3
//
// Pipeline: MLP (f32 WMMA GEMMs) -> feats; A_b = f_b f_b^T + H0 (the
// i*c*sin(outer) term cancels exactly under Hermitianization); eig of H0 once
// (complex Jacobi), per-batch rank-1 secular equation for eigenvalues;
// metrics + controlled output.
//
// GEMM data path: B tiles staged to LDS with GLOBAL_LOAD_ASYNC_TO_LDS_B128
// (ASYNCcnt + s_wait_asynccnt), double-buffered, shared by 4 waves per block.

typedef __attribute__((ext_vector_type(2))) float v2f;
typedef __attribute__((ext_vector_type(8))) float v8f;
typedef int v4i __attribute__((vector_size(16)));   // matches builtin proto

#define BATCH 256
#define DIM   1024
#define HD    512
#define FD    256
#define RD    128

#ifndef __has_builtin
#define __has_builtin(x) 0
#endif
#if __has_builtin(__builtin_amdgcn_global_load_async_to_lds_b128)
#define HAVE_ASYNC_LDS 1
#else
#define HAVE_ASYNC_LDS 0
#endif

// 16-byte global -> LDS copy. Async path when available (ASYNCcnt),
// otherwise plain load/store (completion covered by the barrier).
__device__ __forceinline__ void copy16_to_lds(const float* __restrict__ g,
                                              float* __restrict__ l)
{
#if HAVE_ASYNC_LDS
  __builtin_amdgcn_global_load_async_to_lds_b128(
      (__attribute__((address_space(1))) v4i*)(void*)g,
      (__attribute__((address_space(3))) v4i*)(void*)l,
      0, 0);
#else
  *(float4*)l = *(const float4*)g;
#endif
}

__device__ __forceinline__ void wait_async_copies()
{
#if HAVE_ASYNC_LDS
  asm volatile("s_wait_asynccnt 0" ::: "memory");
#endif
}

// ---------------------------------------------------------------------------
// f32 WMMA GEMM: C[M,N] = A[M,K] * B (+bias) via V_WMMA_F32_16X16X4_F32.
//   BK_NMAJOR=true : Bm is W[N,K] row-major  -> computes A @ W^T
//   BK_NMAJOR=false: Bm is B[K,N] row-major  -> computes A @ B
// Block = 128 threads (4 waves). Each wave owns a 16-row M sub-tile; the
// 16-col B tile (16 x 32 K-chunk) is staged once per block into LDS with
// coalesced b128 async copies and reused by all 4 waves, double-buffered.
// LDS layout: NMAJOR -> [n][k] padded to stride 36 floats (16B-aligned rows,
// lane addresses l15*36 mod 64 all distinct -> conflict-free); K-major -> [k][n].
// A-operand per ISA 7.12.2: lanes 0-15 hold M=lane, K pair selected by lane
// half; C VGPR r -> row (r + 8*half), col = lane&15.
// ---------------------------------------------------------------------------
#define BSTRIDE 36
#define BBUF    (16 * BSTRIDE)   // covers both layouts (16*36 > 32*16)

template<bool BK_NMAJOR>
__device__ __forceinline__
void stage_b_tile(const float* __restrict__ Bm, float* __restrict__ lbuf,
                  int n0, int k0, int N, int K, int t)
{
  if (BK_NMAJOR) {
    const int nl = t >> 3, k4 = (t & 7) << 2;          // 16 rows x 8 float4
    copy16_to_lds(Bm + (size_t)(n0 + nl) * K + k0 + k4, lbuf + nl * BSTRIDE + k4);
  } else {
    const int kl = t >> 2, n4 = (t & 3) << 2;          // 32 rows x 4 float4
    copy16_to_lds(Bm + (size_t)(k0 + kl) * N + n0 + n4, lbuf + kl * 16 + n4);
  }
}

template<bool BK_NMAJOR>
__global__ __launch_bounds__(128)
void wmma_gemm_f32(const float* __restrict__ A, const float* __restrict__ Bm,
                   const float* __restrict__ bias, float* __restrict__ C,
                   int M, int N, int K)
{
  __shared__ float ldsB[2 * BBUF];

  const int t    = threadIdx.x;
  const int lane = t & 31;
  const int wv   = t >> 5;        // wave id 0..3 -> M sub-tile
  const int half = lane >> 4;     // K sub-block: 0 -> k+{0,1}, 1 -> k+{2,3}
  const int l15  = lane & 15;     // A row / B col within the tile
  const int n0 = blockIdx.x * 16;
  const int m0 = blockIdx.y * 64 + wv * 16;

  const float* __restrict__ arow = A + (size_t)(m0 + l15) * K;
  v8f acc = {};

  const int nchunks = K >> 5;     // 32-wide K chunks

  stage_b_tile<BK_NMAJOR>(Bm, ldsB, n0, 0, N, K, t);
  wait_async_copies();
  __syncthreads();

  for (int c = 0; c < nchunks; ++c) {
    const int k0 = c << 5;
    const float* __restrict__ lb = ldsB + (c & 1) * BBUF;
    if (c + 1 < nchunks)
      stage_b_tile<BK_NMAJOR>(Bm, ldsB + ((c + 1) & 1) * BBUF, n0, k0 + 32, N, K, t);

    __builtin_prefetch(arow + k0 + 64, 0, 1);           // -> global_prefetch_b8

#pragma unroll
    for (int kk = 0; kk < 8; ++kk) {
      const int kloc = kk * 4 + 2 * half;
      v2f a, b;
      a.x = arow[k0 + kloc];
      a.y = arow[k0 + kloc + 1];
      if (BK_NMAJOR) {
        b.x = lb[l15 * BSTRIDE + kloc];
        b.y = lb[l15 * BSTRIDE + kloc + 1];
      } else {
        b.x = lb[kloc * 16 + l15];
        b.y = lb[(kloc + 1) * 16 + l15];
      }
      acc = __builtin_amdgcn_wmma_f32_16x16x4_f32(false, a, false, b,
                                                  (short)0, acc, false, false);
    }

    wait_async_copies();
    __syncthreads();
  }

  const float bv = bias ? bias[n0 + l15] : 0.0f;
#pragma unroll
  for (int r = 0; r < 8; ++r) {
    const int row = m0 + r + 8 * half;
    C[(size_t)row * N + n0 + l15] = acc[r] + bv;
  }
}

// ---------------------------------------------------------------------------
// LayerNorm (biased var, eps=1e-5) + ReLU per row of h[B, HD]
// ---------------------------------------------------------------------------
__global__ __launch_bounds__(256)
void ln_relu(const float* __restrict__ h, const float* __restrict__ g,
             const float* __restrict__ be, float* __restrict__ out)
{
  const int b = blockIdx.x;
  const int t = threadIdx.x;
  __shared__ float red[256];
  const float* row = h + (size_t)b * HD;

  float s = 0.f;
  for (int i = t; i < HD; i += 256) s += row[i];
  red[t] = s; __syncthreads();
  for (int o = 128; o > 0; o >>= 1) { if (t < o) red[t] += red[t + o]; __syncthreads(); }
  const float mu = red[0] / HD;
  __syncthreads();

  float v = 0.f;
  for (int i = t; i < HD; i += 256) { const float d = row[i] - mu; v += d * d; }
  red[t] = v; __syncthreads();
  for (int o = 128; o > 0; o >>= 1) { if (t < o) red[t] += red[t + o]; __syncthreads(); }
  const float inv = rsqrtf(red[0] / HD + 1e-5f);

  for (int i = t; i < HD; i += 256) {
    const float y = (row[i] - mu) * inv * g[i] + be[i];
    out[(size_t)b * HD + i] = fmaxf(y, 0.f);
  }
}

// ---------------------------------------------------------------------------
// control = tanh(feats @ W3^T + b3) @ W4^T + b4   -> [B] (written to d_out)
// ---------------------------------------------------------------------------
__global__ __launch_bounds__(128)
void control_head(const float* __restrict__ feats, const float* __restrict__ W3,
                  const float* __restrict__ b3, const float* __restrict__ W4,
                  const float* __restrict__ b4, float* __restrict__ ctl_out)
{
  const int b = blockIdx.x, j = threadIdx.x;
  const float* f = feats + (size_t)b * FD;
  const float* w = W3 + (size_t)j * FD;
  float s = b3[j];
  for (int k = 0; k < FD; ++k) s = fmaf(f[k], w[k], s);
  __shared__ float red[128];
  red[j] = tanhf(s) * W4[j];
  __syncthreads();
  for (int o = 64; o > 0; o >>= 1) { if (j < o) red[j] += red[j + o]; __syncthreads(); }
  if (j == 0) ctl_out[b] = red[0] + b4[0];
}

// ---------------------------------------------------------------------------
// H0 = 0.05*(mem_re + mem_re^T)  +  i * 0.05*(mem_im - mem_im^T)
// ---------------------------------------------------------------------------
__global__ void build_H(const float* __restrict__ mre, const float* __restrict__ mim,
                        float* __restrict__ Hre, float* __restrict__ Him)
{
  const int idx = blockIdx.x * blockDim.x + threadIdx.x;   // i*FD + j
  const int i = idx >> 8, j = idx & (FD - 1);
  Hre[idx] = 0.05f * (mre[i * FD + j] + mre[j * FD + i]);
  Him[idx] = 0.05f * (mim[i * FD + j] - mim[j * FD + i]);
}

// A[b,i,j] = f_b[i]*f_b[j] + Hre[i,j] + i*Him[i,j]  (complex64 interleaved)
__global__ void write_A(const float* __restrict__ feats, const float* __restrict__ Hre,
                        const float* __restrict__ Him, float* __restrict__ Aout)
{
  const int idx = blockIdx.x * blockDim.x + threadIdx.x;   // b*F*F + i*F + j
  const int j = idx & (FD - 1);
  const int i = (idx >> 8) & (FD - 1);
  const int b = idx >> 16;
  const int ij = i * FD + j;
  float2 v;
  v.x = feats[b * FD + i] * feats[b * FD + j] + Hre[ij];
  v.y = Him[ij];
  ((float2*)Aout)[idx] = v;
}

__global__ void init_MQ(const float* __restrict__ Hre, const float* __restrict__ Him,
                        float* __restrict__ Mc, float* __restrict__ Qre,
                        float* __restrict__ Qim)
{
  const int idx = blockIdx.x * blockDim.x + threadIdx.x;
  const int i = idx >> 8, j = idx & (FD - 1);
  Mc[2 * idx]     = Hre[idx];
  Mc[2 * idx + 1] = Him[idx];
  Qre[idx] = (i == j) ? 1.f : 0.f;
  Qim[idx] = 0.f;
}

// ---------------------------------------------------------------------------
// Parallel-order (round-robin) complex Hermitian Jacobi on H0 (one-time).
// One workgroup; 128 disjoint rotations per round; rotation G:
//   G[p][p]=c, G[p][q]=s, G[q][p]=-conj(s), G[q][q]=c,  s = e^{i phi} * t * c
// A <- G^H A G applied as (all column rotations) then (all row rotations).
// ---------------------------------------------------------------------------
__global__ __launch_bounds__(256)
void jacobi_hermitian(float* __restrict__ Mc, float* __restrict__ Qre,
                      float* __restrict__ Qim, float* __restrict__ dvals, int sweeps)
{
  __shared__ int   pp[128], qq[128];
  __shared__ float cc[128], sr[128], si[128];
  const int tid = threadIdx.x;
  const int NM1 = FD - 1;                      // 255 (odd -> perfect matching)

  for (int sw = 0; sw < sweeps; ++sw) {
    for (int r = 0; r < NM1; ++r) {
      if (tid < 128) {
        int p, q;
        if (tid == 0) { p = NM1; q = r; }
        else {
          p = (r + tid) % NM1;
          q = (r - tid + 2 * NM1) % NM1;
        }
        if (p > q) { const int tmp = p; p = q; q = tmp; }
        const float app = Mc[2 * (p * FD + p)];
        const float aqq = Mc[2 * (q * FD + q)];
        const float ar  = Mc[2 * (p * FD + q)];
        const float ai  = Mc[2 * (p * FD + q) + 1];
        const float mag = sqrtf(ar * ar + ai * ai);
        float c = 1.f, s_r = 0.f, s_i = 0.f;
        if (mag > 1e-12f) {
          const float tau = (aqq - app) / (2.f * mag);
          const float tt  = (tau >= 0.f ? 1.f : -1.f)
                          / (fabsf(tau) + sqrtf(1.f + tau * tau));
          c = rsqrtf(1.f + tt * tt);
          const float sm = tt * c;
          s_r = sm * (ar / mag);
          s_i = sm * (ai / mag);
        }
        pp[tid] = p; qq[tid] = q; cc[tid] = c; sr[tid] = s_r; si[tid] = s_i;
      }
      __syncthreads();

      // columns: new_p = c*x - conj(s)*y ; new_q = s*x + c*y   (also Q <- Q*G)
      for (int idx = tid; idx < 128 * FD; idx += 256) {
        const int pr = idx >> 8, row = idx & (FD - 1);
        const int p = pp[pr], q = qq[pr];
        const float c = cc[pr], s_r = sr[pr], s_i = si[pr];
        float xr = Mc[2*(row*FD+p)], xi = Mc[2*(row*FD+p)+1];
        float yr = Mc[2*(row*FD+q)], yi = Mc[2*(row*FD+q)+1];
        Mc[2*(row*FD+p)]   = c*xr - (s_r*yr + s_i*yi);
        Mc[2*(row*FD+p)+1] = c*xi - (s_r*yi - s_i*yr);
        Mc[2*(row*FD+q)]   = (s_r*xr - s_i*xi) + c*yr;
        Mc[2*(row*FD+q)+1] = (s_r*xi + s_i*xr) + c*yi;
        xr = Qre[row*FD+p]; xi = Qim[row*FD+p];
        yr = Qre[row*FD+q]; yi = Qim[row*FD+q];
        Qre[row*FD+p] = c*xr - (s_r*yr + s_i*yi);
        Qim[row*FD+p] = c*xi - (s_r*yi - s_i*yr);
        Qre[row*FD+q] = (s_r*xr - s_i*xi) + c*yr;
        Qim[row*FD+q] = (s_r*xi + s_i*xr) + c*yi;
      }
      __syncthreads();

      // rows: new_p = c*x - s*y ; new_q = conj(s)*x + c*y
      for (int idx = tid; idx < 128 * FD; idx += 256) {
        const int pr = idx >> 8, col = idx & (FD - 1);
        const int p = pp[pr], q = qq[pr];
        const float c = cc[pr], s_r = sr[pr], s_i = si[pr];
        const float xr = Mc[2*(p*FD+col)], xi = Mc[2*(p*FD+col)+1];
        const float yr = Mc[2*(q*FD+col)], yi = Mc[2*(q*FD+col)+1];
        Mc[2*(p*FD+col)]   = c*xr - (s_r*yr - s_i*yi);
        Mc[2*(p*FD+col)+1] = c*xi - (s_r*yi + s_i*yr);
        Mc[2*(q*FD+col)]   = (s_r*xr + s_i*xi) + c*yr;
        Mc[2*(q*FD+col)+1] = (s_r*xi - s_i*xr) + c*yi;
      }
      __syncthreads();
    }
  }
  dvals[tid] = Mc[2 * (tid * FD + tid)];
}

// ---------------------------------------------------------------------------
// Bitonic sort of the 256 H0 eigenvalues, tracking the permutation.
// ---------------------------------------------------------------------------
__global__ __launch_bounds__(256)
void sort_eigs(const float* __restrict__ dvals, float* __restrict__ dsort,
               int* __restrict__ perm)
{
  __shared__ float key[256];
  __shared__ int   idx[256];
  const int t = threadIdx.x;
  key[t] = dvals[t]; idx[t] = t;
  __syncthreads();
  for (int k = 2; k <= 256; k <<= 1) {
    for (int j = k >> 1; j > 0; j >>= 1) {
      const int partner = t ^ j;
      if (partner > t) {
        const bool up = ((t & k) == 0);
        const bool sw = up ? (key[t] > key[partner]) : (key[t] < key[partner]);
        if (sw) {
          const float tk = key[t]; key[t] = key[partner]; key[partner] = tk;
          const int   ti = idx[t]; idx[t] = idx[partner]; idx[partner] = ti;
        }
      }
      __syncthreads();
    }
  }
  dsort[t] = key[t];
  perm[t]  = idx[t];
}

// |z|^2 in sorted order: z = Q^H f  => z2[b][i] = zre^2 + zim^2 at perm[i]
__global__ __launch_bounds__(256)
void z2_perm(const float* __restrict__ zre, const float* __restrict__ zim,
             const int* __restrict__ perm, float* __restrict__ z2)
{
  const int b = blockIdx.x, i = threadIdx.x;
  const int p = perm[i];
  const float a = zre[b * FD + p], c = zim[b * FD + p];
  z2[b * FD + i] = a * a + c * c;
}

// ---------------------------------------------------------------------------
// Secular equation for rank-1 update: eigenvalues of diag(d) + z z^H.
// Root lambda_k in (d_k, d_{k+1}) of  w(l) = 1 + sum z2_i/(d_i - l) = 0
// (w is monotone increasing on each interval) -> bisection, one thread per k.
// ---------------------------------------------------------------------------
__global__ __launch_bounds__(256)
void secular_solve(const float* __restrict__ dsort, const float* __restrict__ z2g,
                   float* __restrict__ ev)
{
  const int b = blockIdx.x, k = threadIdx.x;
  __shared__ float d[256], z[256], red[256];
  d[k] = dsort[k];
  z[k] = z2g[b * FD + k];
  red[k] = z[k];
  __syncthreads();
  for (int o = 128; o > 0; o >>= 1) { if (k < o) red[k] += red[k + o]; __syncthreads(); }
  const float zsum = red[0];

  float lo = d[k];
  float hi = (k == FD - 1) ? d[FD - 1] + zsum + 1e-6f : d[k + 1];
  for (int it = 0; it < 48; ++it) {
    const float mid = 0.5f * (lo + hi);
    float w = 1.f;
    for (int i = 0; i < FD; ++i) {
      float den = d[i] - mid;
      den = (fabsf(den) < 1e-30f) ? copysignf(1e-30f, den) : den;
      w += z[i] / den;
    }
    if (w > 0.f) hi = mid; else lo = mid;
  }
  ev[b * FD + k] = 0.5f * (lo + hi);
}

// metrics per batch: stab = -mean|ev|, det = |prod(ev+1e-6)|, zeros, sigmoid
__global__ __launch_bounds__(256)
void metrics(const float* __restrict__ ev, float* __restrict__ o_est,
             float* __restrict__ o_rzs, float* __restrict__ o_gap,
             float* __restrict__ o_det, float* __restrict__ sig)
{
  const int b = blockIdx.x, t = threadIdx.x;
  __shared__ float sa[256], sp[256];
  const float lam = ev[b * FD + t];
  sa[t] = fabsf(lam);
  sp[t] = lam + 1e-6f;
  __syncthreads();
  for (int o = 128; o > 0; o >>= 1) {
    if (t < o) { sa[t] += sa[t + o]; sp[t] *= sp[t + o]; }
    __syncthreads();
  }
  if (t == 0) {
    const float stab = -sa[0] / FD;
    o_est[b] = stab;
    o_rzs[b] = 0.f;                 // exactly zero for Hermitian A
    o_gap[b] = 0.f;                 // exactly zero for Hermitian A
    o_det[b] = fabsf(sp[0]);
    sig[b] = 1.f / (1.f + expf(-stab));
  }
}

__global__ void apply_control(const float* __restrict__ x, const float* __restrict__ sig,
                              float* __restrict__ out)
{
  const int idx = blockIdx.x * blockDim.x + threadIdx.x;
  const int b = idx >> 10;          // DIM = 1024
  out[idx] = x[idx] * (1.f + 0.1f * sig[b]);
}

// ---------------------------------------------------------------------------
extern "C" void kernel_launch(void* const* d_in, const int* in_sizes, int n_in,
                              void* d_out, int out_size, void* d_ws, size_t ws_size,
                              hipStream_t stream)
{
  (void)in_sizes; (void)n_in; (void)out_size; (void)ws_size;

  const float* x   = (const float*)d_in[0];
  const float* W1  = (const float*)d_in[1];
  const float* b1  = (const float*)d_in[2];
  const float* g1  = (const float*)d_in[3];
  const float* be1 = (const float*)d_in[4];
  const float* W2  = (const float*)d_in[5];
  const float* b2  = (const float*)d_in[6];
  const float* W3  = (const float*)d_in[7];
  const float* b3  = (const float*)d_in[8];
  const float* W4  = (const float*)d_in[9];
  const float* b4  = (const float*)d_in[10];
  const float* mre = (const float*)d_in[11];
  const float* mim = (const float*)d_in[12];

  // outputs, concatenated flat in return order (complex64 -> float pairs)
  float* out    = (float*)d_out;
  float* o_ctrl = out;                                   // [B, D]
  float* o_A    = out + (size_t)BATCH * DIM;             // [B, F, F] cplx
  float* o_est  = o_A + (size_t)BATCH * FD * FD * 2;     // [B]
  float* o_rzs  = o_est + BATCH;                         // [B]
  float* o_gap  = o_rzs + BATCH;                         // [B]
  float* o_det  = o_gap + BATCH;                         // [B]
  float* o_con  = o_det + BATCH;                         // [B, 1]

  // workspace carve-out (~4 MB of floats)
  float* W = (float*)d_ws;
  size_t off = 0;
  auto alloc = [&](size_t n) { float* p = W + off; off += n; return p; };
  float* h     = alloc((size_t)BATCH * HD);
  float* hn    = alloc((size_t)BATCH * HD);
  float* feats = alloc((size_t)BATCH * FD);
  float* Hre   = alloc((size_t)FD * FD);
  float* Him   = alloc((size_t)FD * FD);
  float* Mc    = alloc((size_t)2 * FD * FD);
  float* Qre   = alloc((size_t)FD * FD);
  float* Qim   = alloc((size_t)FD * FD);
  float* zre   = alloc((size_t)BATCH * FD);
  float* zim   = alloc((size_t)BATCH * FD);
  float* z2    = alloc((size_t)BATCH * FD);
  float* dvals = alloc(FD);
  float* dsort = alloc(FD);
  int*   perm  = (int*)alloc(FD);
  float* ev    = alloc((size_t)BATCH * FD);
  float* sig   = alloc(BATCH);

  // 1) h = x @ W1^T + b1            [256,512]  (WMMA f32, async-LDS B tiles)
  wmma_gemm_f32<true><<<dim3(HD / 16, BATCH / 64), 128, 0, stream>>>(
      x, W1, b1, h, BATCH, HD, DIM);
  // 2) LayerNorm + ReLU
  ln_relu<<<BATCH, 256, 0, stream>>>(h, g1, be1, hn);
  // 3) feats = hn @ W2^T + b2       [256,256]  (WMMA f32)
  wmma_gemm_f32<true><<<dim3(FD / 16, BATCH / 64), 128, 0, stream>>>(
      hn, W2, b2, feats, BATCH, FD, HD);
  // 4) control head -> d_out (not needed downstream: sin term cancels)
  control_head<<<BATCH, RD, 0, stream>>>(feats, W3, b3, W4, b4, o_con);
  // 5) constant Hermitian part H0
  build_H<<<(FD * FD) / 256, 256, 0, stream>>>(mre, mim, Hre, Him);
  // 6) stream A to d_out (134 MB, float2 coalesced stores, L2-resident)
  write_A<<<(BATCH * FD * FD) / 256, 256, 0, stream>>>(feats, Hre, Him, o_A);
  // 7) one-time eigendecomposition of H0
  init_MQ<<<(FD * FD) / 256, 256, 0, stream>>>(Hre, Him, Mc, Qre, Qim);
  jacobi_hermitian<<<1, 256, 0, stream>>>(Mc, Qre, Qim, dvals, 10);
  sort_eigs<<<1, 256, 0, stream>>>(dvals, dsort, perm);
  // 8) z = Q^H feats_b  (two real WMMA GEMMs; sign of zim irrelevant for |z|^2)
  wmma_gemm_f32<false><<<dim3(FD / 16, BATCH / 64), 128, 0, stream>>>(
      feats, Qre, nullptr, zre, BATCH, FD, FD);
  wmma_gemm_f32<false><<<dim3(FD / 16, BATCH / 64), 128, 0, stream>>>(
      feats, Qim, nullptr, zim, BATCH, FD, FD);
  z2_perm<<<BATCH, FD, 0, stream>>>(zre, zim, perm, z2);
  // 9) per-batch eigenvalues via secular equation
  secular_solve<<<BATCH, FD, 0, stream>>>(dsort, z2, ev);
  // 10) metrics + 11) controlled output
  metrics<<<BATCH, FD, 0, stream>>>(ev, o_est, o_rzs, o_gap, o_det, sig);
  apply_control<<<(BATCH * DIM) / 256, 256, 0, stream>>>(x, sig, o_ctrl);
}